// AttnReweight_76647986364468
// MI455X (gfx1250) — compile-verified
//
#include <hip/hip_runtime.h>
#include <hip/hip_bf16.h>
#include <math.h>

// Shapes fixed by setup_inputs(): B=2, HD=4, H=W=256, K=49 (7x7), sH=sW=32, 8x8 pixels/superpixel.
#define EPSV 1e-15f

#if __has_builtin(__builtin_amdgcn_global_load_async_to_lds_b32)
#define HAS_ASYNC_LDS 1
#else
#define HAS_ASYNC_LDS 0
#endif

typedef __attribute__((address_space(1))) int* gas_iptr;  // global (AS1) int*
typedef __attribute__((address_space(3))) int* las_iptr;  // LDS    (AS3) int*

__launch_bounds__(256)
__global__ void AttnReweight_76647986364468_kernel(const float* __restrict__ attn,
                                                   const float* __restrict__ sims,
                                                   float* __restrict__ out) {
    const int b   = blockIdx.z;   // batch
    const int a   = blockIdx.y;   // superpixel row  == h/8 for all queries in tile
    const int c   = blockIdx.x;   // superpixel col  == w/8
    const int tid = threadIdx.x;  // 0..255

    __shared__ float S[196 * 9];   // [j: 14x14 key region][s: 9 superpixel neighbors]
    __shared__ float Wt[64 * 49];  // [q: 8x8 queries][k: 49 window offsets]

    // ---------- Phase 1: gather sims into LDS via CDNA5 async-to-LDS ----------
    for (int e = tid; e < 196 * 9; e += 256) {
        int j  = e / 9;
        int s  = e - 9 * j;
        int jh = a * 8 + (j / 14) - 3;           // key pixel row, region [-3, +10]
        int jw = c * 8 + (j % 14) - 3;
        jh = jh < 0 ? 0 : (jh > 255 ? 255 : jh); // clip like reference
        jw = jw < 0 ? 0 : (jw > 255 ? 255 : jw);
        int dh = s / 3 - 1;
        int dw = s - 3 * (s / 3) - 1;
        int sphc = a + dh; sphc = sphc < 0 ? 0 : (sphc > 31 ? 31 : sphc);
        int spwc = c + dw; spwc = spwc < 0 ? 0 : (spwc > 31 ? 31 : spwc);
        size_t g = ((((size_t)b * 256 + (size_t)jh) * 256 + (size_t)jw) << 10)
                   + (size_t)(sphc * 32 + spwc);
#if HAS_ASYNC_LDS
        __builtin_amdgcn_global_load_async_to_lds_b32(
            (gas_iptr)(sims + g), (las_iptr)(&S[e]), 0, 0);
#else
        S[e] = sims[g];
#endif
    }
#if HAS_ASYNC_LDS
 #if __has_builtin(__builtin_amdgcn_s_wait_asynccnt)
    __builtin_amdgcn_s_wait_asynccnt(0);
 #else
    asm volatile("s_wait_asynccnt 0" ::: "memory");
 #endif
#endif
    __syncthreads();

    // ---------- Phase 2: superpixel weights Wt[q][k] = sum_s Pi[s] * S[j(q,k)][s] ----------
    {
        int q    = tid >> 2;                    // 0..63
        int part = tid & 3;                     // split K among 4 threads
        int k0   = (part * 49) >> 2;
        int k1   = ((part + 1) * 49) >> 2;
        int hq   = q >> 3, wq = q & 7;
        const float* Sc = &S[((hq + 3) * 14 + (wq + 3)) * 9];  // query pixel's own slot
        float Pi[9];
#pragma unroll
        for (int s = 0; s < 9; ++s) {
            int dh = s / 3 - 1, dw = s % 3 - 1;
            bool valid = ((unsigned)(a + dh) < 32u) && ((unsigned)(c + dw) < 32u);
            Pi[s] = valid ? Sc[s] : 0.0f;       // mask once; product kills Pj side too
        }
        for (int k = k0; k < k1; ++k) {
            int kh = k / 7, kw = k - 7 * kh;    // 0..6 (offset -3..+3 folded into +3 region base)
            const float* Sj = &S[((hq + kh) * 14 + (wq + kw)) * 9];
            float acc = 0.0f;
#pragma unroll
            for (int s = 0; s < 9; ++s) acc = fmaf(Pi[s], Sj[s], acc);
            Wt[q * 49 + k] = acc;
        }
    }
    __syncthreads();

    // ---------- Phase 3: masked softmax reweight, one thread per (head, query) ----------
    {
        int q  = tid & 63;
        int hd = tid >> 6;                      // 0..3
        int h  = a * 8 + (q >> 3);
        int w  = c * 8 + (q & 7);
        size_t base = ((((size_t)b * 4 + (size_t)hd) * 256 + (size_t)h) * 256 + (size_t)w) * 49;
        const float* arow = attn + base;

        float av[49];
        float m = -INFINITY;
#pragma unroll
        for (int k = 0; k < 49; ++k) {
            av[k] = __builtin_nontemporal_load(arow + k);   // streamed once: keep sims in L2
            m = fmaxf(m, av[k]);
        }
        const float* wrow = &Wt[q * 49];
        float ssum = 0.0f;
#pragma unroll
        for (int k = 0; k < 49; ++k) {
            float t = wrow[k] * expf(av[k] - m);
            av[k] = t;
            ssum += t;
        }
        float inv = 1.0f / (EPSV + ssum);
        float* orow = out + base;
#pragma unroll
        for (int k = 0; k < 49; ++k) {
            __builtin_nontemporal_store(av[k] * inv, orow + k);
        }
    }
}

extern "C" void kernel_launch(void* const* d_in, const int* in_sizes, int n_in,
                              void* d_out, int out_size, void* d_ws, size_t ws_size,
                              hipStream_t stream) {
    const float* attn = (const float*)d_in[0];  // (2,4,256,256,49) f32
    const float* sims = (const float*)d_in[1];  // (2,256,256,32,32) f32
    float* out = (float*)d_out;                 // (2,4,256,256,49) f32
    (void)in_sizes; (void)n_in; (void)out_size; (void)d_ws; (void)ws_size;

    dim3 grid(32, 32, 2);   // (c tiles, a tiles, batch) -> 2048 blocks
    dim3 block(256);        // 8 waves (wave32)
    AttnReweight_76647986364468_kernel<<<grid, block, 0, stream>>>(attn, sims, out);
}